// PretrainedCorrespondence_88914412962141
// MI455X (gfx1250) — compile-verified
//
#include <hip/hip_runtime.h>
#include <stdint.h>

// ---------------------------------------------------------------------------
// PretrainedCorrespondence for MI455X (gfx1250, wave32)
//   B=16, C=384, FS=56, hw=3136, IMG=448, k=512
// Pipeline:
//   1) bilinear mask downsample 448->56, threshold >0.5          (smd, tmd)
//   2) fused fp32 WMMA GEMM (src^T @ tgt) + mask + dual argmax   (fwp, bwp)
//      - V_WMMA_F32_16X16X4_F32, 64x64 block tile, 4 accumulators/wave
//      - LDS in k-contiguous padded layout -> ds_load_b64 fragments
//      - argmax via packed (ordered_f32<<32 | ~idx) u64 atomicMax
//   3) cycle distance  dist[t] = ||grid[:,fw[bw[t]]] - grid[:,t]||
//   4) per-batch top-512 (sequential min-selection, LDS resident)
//   5) output gathers
// ---------------------------------------------------------------------------

typedef __attribute__((ext_vector_type(2))) float v2f;
typedef __attribute__((ext_vector_type(8))) float v8f;

#define B_   16
#define C_   384
#define FS_  56
#define HW_  (FS_ * FS_)   // 3136
#define IMG_ 448
#define KSEL 512

#define TM 64        // block src tile (M)
#define TN 64        // block tgt tile (N)
#define KC 64        // K chunk staged in LDS
#define KCP (KC + 2) // padded LDS row stride (floats), keeps b64 alignment
#define NTHREADS 128 // 4 waves

// monotonic float -> uint mapping for unsigned max compare
__device__ __forceinline__ unsigned int ordf(float f) {
  unsigned int u = __float_as_uint(f);
  return (u & 0x80000000u) ? ~u : (u | 0x80000000u);
}

// --------------------------- 1) mask downsample ----------------------------
__global__ void mask_down_kernel(const float* __restrict__ mask,
                                 float* __restrict__ md) {
  int idx = blockIdx.x * blockDim.x + threadIdx.x;
  if (idx >= B_ * HW_) return;
  int b = idx / HW_;
  int p = idx % HW_;
  int y = p / FS_, x = p % FS_;
  float cy = (y + 0.5f) * 8.0f - 0.5f;
  float cx = (x + 0.5f) * 8.0f - 0.5f;
  cy = fminf(fmaxf(cy, 0.0f), (float)(IMG_ - 1));
  cx = fminf(fmaxf(cx, 0.0f), (float)(IMG_ - 1));
  int y0 = (int)floorf(cy), x0 = (int)floorf(cx);
  int y1 = min(y0 + 1, IMG_ - 1), x1 = min(x0 + 1, IMG_ - 1);
  float wy = cy - (float)y0, wx = cx - (float)x0;
  const float* mb = mask + (size_t)b * IMG_ * IMG_;
  float v00 = mb[(size_t)y0 * IMG_ + x0], v01 = mb[(size_t)y0 * IMG_ + x1];
  float v10 = mb[(size_t)y1 * IMG_ + x0], v11 = mb[(size_t)y1 * IMG_ + x1];
  float v = v00 * (1.0f - wy) * (1.0f - wx) + v01 * (1.0f - wy) * wx +
            v10 * wy * (1.0f - wx) + v11 * wy * wx;
  md[idx] = (v > 0.5f) ? 1.0f : 0.0f;
}

// --------------------------- init argmax keys ------------------------------
__global__ void init_kernel(unsigned long long* __restrict__ fwp,
                            unsigned long long* __restrict__ bwp) {
  int idx = blockIdx.x * blockDim.x + threadIdx.x;
  if (idx >= B_ * HW_) return;
  fwp[idx] = 0ull;
  bwp[idx] = 0ull;
}

// ------------------ 2) fused WMMA correlation + dual argmax ----------------
__global__ __launch_bounds__(NTHREADS)
void corr_argmax_kernel(const float* __restrict__ src,
                        const float* __restrict__ tgt,
                        const float* __restrict__ smd,
                        const float* __restrict__ tmd,
                        unsigned long long* __restrict__ fwp,
                        unsigned long long* __restrict__ bwp) {
  // k-contiguous, padded: row r (src row m / tgt col n), element k at [r*KCP+k]
  __shared__ __align__(16) float As[TM * KCP];  // 16.9 KB
  __shared__ __align__(16) float Bs[TN * KCP];  // 16.9 KB
  const int b   = blockIdx.z;
  const int s0  = blockIdx.x * TM;
  const int t0g = blockIdx.y * TN;
  const int tid  = threadIdx.x;
  const int lane = tid & 31;
  const int wave = tid >> 5;                // 0..3: M rows [wave*16, wave*16+16)
  const float* srcb = src + (size_t)b * C_ * HW_;
  const float* tgtb = tgt + (size_t)b * C_ * HW_;

  v8f acc[4] = {};                          // 4 N-chunks of 16 cols each

  const int mloc = lane & 15;               // M (A) / N (B) position in half-wave
  const int half = lane >> 4;
  const int koff = half * 2;                // lanes 0-15: K={0,1}; 16-31: K={2,3}

  for (int k0 = 0; k0 < C_; k0 += KC) {
    // global -> LDS staging, float4 (b128) loads along the spatial axis
    for (int i = tid; i < TM * KC / 4; i += NTHREADS) {
      int k = i >> 4;               // 16 float4 per k-row
      int m = (i & 15) * 4;
      float4 v = *(const float4*)&srcb[(size_t)(k0 + k) * HW_ + s0 + m];
      As[(m + 0) * KCP + k] = v.x;
      As[(m + 1) * KCP + k] = v.y;
      As[(m + 2) * KCP + k] = v.z;
      As[(m + 3) * KCP + k] = v.w;
    }
    for (int i = tid; i < TN * KC / 4; i += NTHREADS) {
      int k = i >> 4;
      int n = (i & 15) * 4;
      float4 v = *(const float4*)&tgtb[(size_t)(k0 + k) * HW_ + t0g + n];
      Bs[(n + 0) * KCP + k] = v.x;
      Bs[(n + 1) * KCP + k] = v.y;
      Bs[(n + 2) * KCP + k] = v.z;
      Bs[(n + 3) * KCP + k] = v.w;
    }
    __syncthreads();

    const float* Arow = &As[(wave * 16 + mloc) * KCP];
#pragma unroll
    for (int kk = 0; kk < KC; kk += 4) {
      v2f a = *(const v2f*)&Arow[kk + koff];      // 8B aligned -> ds_load_b64
#pragma unroll
      for (int j = 0; j < 4; ++j) {
        v2f bb = *(const v2f*)&Bs[(j * 16 + mloc) * KCP + kk + koff];
        acc[j] = __builtin_amdgcn_wmma_f32_16x16x4_f32(
            /*neg_a=*/false, a, /*neg_b=*/false, bb,
            /*c_mod=*/(short)0, acc[j], /*reuse_a=*/false, /*reuse_b=*/false);
      }
    }
    __syncthreads();
  }

  // ----- mask + in-wave argmax reductions -----
  unsigned long long* fwpb = fwp + (size_t)b * HW_;
  unsigned long long* bwpb = bwp + (size_t)b * HW_;

  float tmv[4];
#pragma unroll
  for (int j = 0; j < 4; ++j)
    tmv[j] = tmd[(size_t)b * HW_ + t0g + j * 16 + mloc];

  float cb[4];  int cbi[4];                 // per-N-chunk column running max
#pragma unroll
  for (int j = 0; j < 4; ++j) { cb[j] = -3.0e38f; cbi[j] = 0; }

#pragma unroll
  for (int r = 0; r < 8; ++r) {
    const int s = s0 + wave * 16 + r + half * 8;   // C/D rows M=r / r+8 by half
    const float sm = smd[(size_t)b * HW_ + s];

    // fold the 4 N-chunks in-lane (ascending t -> first-occurrence ties)
    float rv = -3.0e38f;
    int   ri = 0;
#pragma unroll
    for (int j = 0; j < 4; ++j) {
      const float v = (sm > 0.0f && tmv[j] > 0.0f) ? acc[j][r] : -100000.0f;
      const int   t = t0g + j * 16 + mloc;
      if (v > rv || (v == rv && t < ri)) { rv = v; ri = t; }
      if (v > cb[j] || (v == cb[j] && s < cbi[j])) { cb[j] = v; cbi[j] = s; }
    }
    // fw[s]: reduce over the 16 lanes of this half
#pragma unroll
    for (int mk = 1; mk < 16; mk <<= 1) {
      float ov = __shfl_xor(rv, mk, 32);
      int   oi = __shfl_xor(ri, mk, 32);
      if (ov > rv || (ov == rv && oi < ri)) { rv = ov; ri = oi; }
    }
    if (mloc == 0) {
      unsigned long long key =
          ((unsigned long long)ordf(rv) << 32) | (unsigned int)(~(unsigned int)ri);
      atomicMax(&fwpb[s], key);
    }
  }

  // bw[t]: merge halves (lane l <-> l+16 share the same N), one atomic per col
#pragma unroll
  for (int j = 0; j < 4; ++j) {
    float ov = __shfl_xor(cb[j], 16, 32);
    int   oi = __shfl_xor(cbi[j], 16, 32);
    if (ov > cb[j] || (ov == cb[j] && oi < cbi[j])) { cb[j] = ov; cbi[j] = oi; }
    if (half == 0) {
      unsigned long long key =
          ((unsigned long long)ordf(cb[j]) << 32) |
          (unsigned int)(~(unsigned int)cbi[j]);
      atomicMax(&bwpb[t0g + j * 16 + mloc], key);
    }
  }
}

// ------------------------- 3) cycle distance -------------------------------
__global__ void distance_kernel(const unsigned long long* __restrict__ fwp,
                                const unsigned long long* __restrict__ bwp,
                                const float* __restrict__ grid,
                                const float* __restrict__ tmd,
                                float* __restrict__ dist) {
  int idx = blockIdx.x * blockDim.x + threadIdx.x;
  if (idx >= B_ * HW_) return;
  int b = idx / HW_, t = idx % HW_;
  int bw = (int)(~(unsigned int)bwp[idx]);                   // best src for t
  int cy = (int)(~(unsigned int)fwp[(size_t)b * HW_ + bw]);  // best tgt for bw
  const float* gb = grid + (size_t)b * 2 * HW_;
  float dx = gb[cy] - gb[t];
  float dy = gb[HW_ + cy] - gb[HW_ + t];
  float d = sqrtf(dx * dx + dy * dy);
  dist[idx] = (tmd[idx] > 0.0f) ? d : 100000.0f;
}

// ------------------------- 4) per-batch top-k ------------------------------
__global__ __launch_bounds__(256)
void topk_kernel(const float* __restrict__ dist, int* __restrict__ idxout) {
  __shared__ float ld[HW_];
  __shared__ unsigned long long red[256];
  const int b = blockIdx.x, t = threadIdx.x;
  for (int i = t; i < HW_; i += 256) ld[i] = dist[(size_t)b * HW_ + i];
  __syncthreads();
  for (int r = 0; r < KSEL; ++r) {
    unsigned long long best = 0xFFFFFFFFFFFFFFFFull;
    for (int i = t; i < HW_; i += 256) {
      unsigned long long key =
          ((unsigned long long)ordf(ld[i]) << 32) | (unsigned int)i;
      best = (key < best) ? key : best;
    }
    red[t] = best;
    __syncthreads();
    for (int s = 128; s > 0; s >>= 1) {
      if (t < s) red[t] = (red[t + s] < red[t]) ? red[t + s] : red[t];
      __syncthreads();
    }
    if (t == 0) {
      int sel = (int)(unsigned int)red[0];
      idxout[b * KSEL + r] = sel;
      ld[sel] = 3.0e38f;  // remove from further rounds
    }
    __syncthreads();
  }
}

// ----------------------------- 5) gathers ----------------------------------
__global__ void gather_kernel(const int* __restrict__ idxs,
                              const unsigned long long* __restrict__ bwp,
                              const float* __restrict__ grid,
                              const float* __restrict__ tmd,
                              float* __restrict__ out) {
  int j = blockIdx.x * blockDim.x + threadIdx.x;
  if (j >= B_ * KSEL) return;
  int b = j / KSEL, q = j % KSEL;
  int i = idxs[j];
  int bwi = (int)(~(unsigned int)bwp[(size_t)b * HW_ + i]);
  const float* gb = grid + (size_t)b * 2 * HW_;
  float* match = out;                           // [B,2,k]
  float* gout  = out + B_ * 2 * KSEL;           // [B,2,k]
  float* imat  = out + 2 * B_ * 2 * KSEL;       // [B,k]
  float* iout  = imat + B_ * KSEL;              // [B,k]
  float* mm    = iout + B_ * KSEL;              // [B,k]
  match[(b * 2 + 0) * KSEL + q] = gb[bwi];
  match[(b * 2 + 1) * KSEL + q] = gb[HW_ + bwi];
  gout[(b * 2 + 0) * KSEL + q]  = gb[i];
  gout[(b * 2 + 1) * KSEL + q]  = gb[HW_ + i];
  imat[j] = (float)bwi;
  iout[j] = (float)i;
  mm[j]   = tmd[(size_t)b * HW_ + i];
}

// ---------------------------------------------------------------------------
extern "C" void kernel_launch(void* const* d_in, const int* in_sizes, int n_in,
                              void* d_out, int out_size, void* d_ws, size_t ws_size,
                              hipStream_t stream) {
  const float* src   = (const float*)d_in[0];   // [B,C,hw]
  const float* tgt   = (const float*)d_in[1];   // [B,C,hw]
  const float* smask = (const float*)d_in[2];   // [B,IMG,IMG]
  const float* tmask = (const float*)d_in[3];   // [B,IMG,IMG]
  const float* grid  = (const float*)d_in[4];   // [B,2,FS,FS]
  // d_in[5] = k (fixed 512)

  char* ws = (char*)d_ws;
  const size_t n = (size_t)B_ * HW_;
  float* smd = (float*)ws;                           ws += n * sizeof(float);
  float* tmd = (float*)ws;                           ws += n * sizeof(float);
  unsigned long long* fwp = (unsigned long long*)ws; ws += n * sizeof(unsigned long long);
  unsigned long long* bwp = (unsigned long long*)ws; ws += n * sizeof(unsigned long long);
  float* dist = (float*)ws;                          ws += n * sizeof(float);
  int* idxs = (int*)ws;

  float* out = (float*)d_out;

  const int blocksN = (int)((n + 255) / 256);
  mask_down_kernel<<<blocksN, 256, 0, stream>>>(smask, smd);
  mask_down_kernel<<<blocksN, 256, 0, stream>>>(tmask, tmd);
  init_kernel<<<blocksN, 256, 0, stream>>>(fwp, bwp);

  dim3 g(HW_ / TM, HW_ / TN, B_);   // 49 x 49 x 16
  corr_argmax_kernel<<<g, NTHREADS, 0, stream>>>(src, tgt, smd, tmd, fwp, bwp);

  distance_kernel<<<blocksN, 256, 0, stream>>>(fwp, bwp, grid, tmd, dist);
  topk_kernel<<<B_, 256, 0, stream>>>(dist, idxs);
  gather_kernel<<<(B_ * KSEL + 255) / 256, 256, 0, stream>>>(idxs, bwp, grid, tmd, out);
}